// MPDEncoder_49297634624101
// MI455X (gfx1250) — compile-verified
//
#include <hip/hip_runtime.h>

typedef __attribute__((ext_vector_type(2))) float v2f;
typedef __attribute__((ext_vector_type(8))) float v8f;

#define B_N 1024
#define F_INN 256
#define N_DD 64
#define N_AA 64
#define VBS 128
#define N_STEPS 3
#define FO 128
#define D_CAT 512
#define AD 256
#define GAMMA_C 1.3f
#define EPS_M 1e-15f
#define BN_EPS 1e-5f
#define SQRT_HALF_C 0.70710678118654752f
#define GEMM_M 1024
#define GEMM_N 256

// ---------------- full-batch batchnorm: one block per feature ----------------
__global__ void bn_kernel(const float* __restrict__ x, const float* __restrict__ w,
                          const float* __restrict__ b, float* __restrict__ xn) {
    int f = blockIdx.x, t = threadIdx.x;
    __shared__ float rs[256], rs2[256];
    float s = 0.f, s2 = 0.f;
    for (int r = t; r < B_N; r += 256) {
        float v = x[(size_t)r * F_INN + f];
        s += v; s2 += v * v;
    }
    rs[t] = s; rs2[t] = s2; __syncthreads();
    for (int off = 128; off > 0; off >>= 1) {
        if (t < off) { rs[t] += rs[t + off]; rs2[t] += rs2[t + off]; }
        __syncthreads();
    }
    float m = rs[0] * (1.f / B_N);
    float var = rs2[0] * (1.f / B_N) - m * m;
    float scl = rsqrtf(var + BN_EPS) * w[f];
    float bb = b[f] - m * scl;
    for (int r = t; r < B_N; r += 256)
        xn[(size_t)r * F_INN + f] = x[(size_t)r * F_INN + f] * scl + bb;
}

// ---------------- concat [a(256), b(256)] -> out(512) ----------------
__global__ void concat_kernel(const float* __restrict__ a, const float* __restrict__ b,
                              float* __restrict__ out) {
    int i = blockIdx.x * 256 + threadIdx.x;   // B*512 total
    int r = i >> 9, c = i & 511;
    out[i] = (c < 256) ? a[r * 256 + c] : b[r * 256 + (c - 256)];
}

// ---------------- WMMA f32 GEMM: D(M x 256) = A(M x K) * B ----------------
// TRANSB=1: B given as W(N,K) row-major => B[k][n] = W[n*K+k]   (x @ W.T)
// TRANSB=0: B given as (K,N) row-major  => B[k][n] = W[k*N+n]   (mask @ gmat)
// one wave computes one 16x16 tile of D via V_WMMA_F32_16X16X4_F32
template <bool TRANSB>
__global__ void gemm_wmma(const float* __restrict__ A, const float* __restrict__ Bm,
                          float* __restrict__ D, int K) {
    const int lane = threadIdx.x & 31;
    const int wave = threadIdx.x >> 5;        // 4 waves per block
    const int tm = blockIdx.x;                // M/16 tiles
    const int tn = blockIdx.y * 4 + wave;     // N/16 tiles
    const int r16 = lane & 15;
    const int half = lane >> 4;               // 0 or 1
    const int row = tm * 16 + r16;            // A row owned by this lane
    const int col = tn * 16 + r16;            // B col owned by this lane
    const int kh = half * 2;                  // K sub-offset per ISA A/B layout
    v8f acc = {};
    for (int kb = 0; kb < K; kb += 4) {
        v2f av, bv;
        const float* ap = A + (size_t)row * K + kb + kh;
        av.x = ap[0]; av.y = ap[1];
        if (TRANSB) {
            const float* bp = Bm + (size_t)col * K + kb + kh;
            bv.x = bp[0]; bv.y = bp[1];
        } else {
            const float* bp = Bm + (size_t)(kb + kh) * GEMM_N + col;
            bv.x = bp[0]; bv.y = bp[GEMM_N];
        }
        acc = __builtin_amdgcn_wmma_f32_16x16x4_f32(false, av, false, bv,
                                                    (short)0, acc, false, false);
    }
    const int mbase = tm * 16 + half * 8;     // VGPR r -> M = mbase + r
#pragma unroll
    for (int r = 0; r < 8; ++r)
        D[(size_t)(mbase + r) * GEMM_N + col] = acc[r];
}

// ---------------- ghost-BN (groups of VBS rows, 256 features), no GLU ----------------
__global__ void gbn_kernel(const float* __restrict__ h, const float* __restrict__ w,
                           const float* __restrict__ b, float* __restrict__ out) {
    __shared__ float soff[256], sscl[256];
    int g = blockIdx.x, f = threadIdx.x;
    const float* hp = h + (size_t)g * VBS * 256;
    float s = 0.f, s2 = 0.f;
    for (int r = 0; r < VBS; ++r) { float v = hp[r * 256 + f]; s += v; s2 += v * v; }
    float m = s * (1.f / VBS);
    float var = s2 * (1.f / VBS) - m * m;
    float scl = rsqrtf(var + BN_EPS) * w[f];
    sscl[f] = scl; soff[f] = b[f] - m * scl;
    __syncthreads();
    float* op = out + (size_t)g * VBS * 256;
    for (int i = f; i < VBS * 256; i += 256) {
        int c = i & 255;
        op[i] = hp[i] * sscl[c] + soff[c];
    }
}

// ---------------- ghost-BN + GLU: h (B x 256) -> out (B x 128) ----------------
__global__ void gbn_glu_kernel(const float* __restrict__ h, const float* __restrict__ w,
                               const float* __restrict__ b, float* __restrict__ out) {
    __shared__ float soff[256], sscl[256];
    int g = blockIdx.x, f = threadIdx.x;
    const float* hp = h + (size_t)g * VBS * 256;
    float s = 0.f, s2 = 0.f;
    for (int r = 0; r < VBS; ++r) { float v = hp[r * 256 + f]; s += v; s2 += v * v; }
    float m = s * (1.f / VBS);
    float var = s2 * (1.f / VBS) - m * m;
    float scl = rsqrtf(var + BN_EPS) * w[f];
    sscl[f] = scl; soff[f] = b[f] - m * scl;
    __syncthreads();
    float* op = out + (size_t)g * VBS * 128;
    for (int i = f; i < VBS * 128; i += 256) {
        int r = i >> 7, j = i & 127;
        float a0 = hp[r * 256 + j] * sscl[j] + soff[j];
        float a1 = hp[r * 256 + j + 128] * sscl[j + 128] + soff[j + 128];
        op[i] = a0 * (1.f / (1.f + __expf(-a1)));
    }
}

// ---------------- (a + b) * sqrt(0.5) ----------------
__global__ void residual_kernel(const float* __restrict__ a, const float* __restrict__ b,
                                float* __restrict__ out) {
    int i = blockIdx.x * 256 + threadIdx.x;   // B*128
    out[i] = (a[i] + b[i]) * SQRT_HALF_C;
}

// ---------------- feat (B x 128) -> steps(leaky) + att ----------------
__global__ void split_feat_kernel(const float* __restrict__ feat, float* __restrict__ step_out,
                                  float* __restrict__ att, int write_steps) {
    int i = blockIdx.x * 256 + threadIdx.x;   // B*128
    int b = i >> 7, j = i & 127;
    float v = feat[i];
    if (j < 64) {
        if (write_steps) step_out[b * 64 + j] = (v > 0.f) ? v : 0.01f * v;
    } else {
        att[b * 64 + (j - 64)] = v;
    }
}

// ---------------- elementwise helpers ----------------
__global__ void mul_kernel(const float* __restrict__ a, const float* __restrict__ b,
                           float* __restrict__ out) {
    int i = blockIdx.x * 256 + threadIdx.x;
    out[i] = a[i] * b[i];
}
__global__ void fill_kernel(float* __restrict__ p, float v, int n) {
    int i = blockIdx.x * 256 + threadIdx.x;
    if (i < n) p[i] = v;
}
__global__ void finalize_loss_kernel(const float* __restrict__ acc, float* __restrict__ out) {
    out[0] = acc[0] * (1.f / ((float)B_N * (float)N_STEPS));
}

// ---------------- sparsemax over 256 features, one block per row ----------------
__global__ void sparsemax_kernel(const float* __restrict__ a, float* __restrict__ prior,
                                 float* __restrict__ mask, float* __restrict__ loss_acc) {
    int row = blockIdx.x, t = threadIdx.x;
    __shared__ float zs[256], zc[256], red[256];
    __shared__ int kz_s;
    float pr = prior[(size_t)row * 256 + t];
    float z = a[(size_t)row * 256 + t] * pr;
    zs[t] = z;
    __syncthreads();
    // bitonic sort, descending
    for (int k = 2; k <= 256; k <<= 1) {
        for (int j = k >> 1; j > 0; j >>= 1) {
            int ixj = t ^ j;
            if (ixj > t) {
                float a0 = zs[t], a1 = zs[ixj];
                bool desc = ((t & k) == 0);
                if (desc ? (a0 < a1) : (a0 > a1)) { zs[t] = a1; zs[ixj] = a0; }
            }
            __syncthreads();
        }
    }
    // inclusive prefix sum of sorted values
    zc[t] = zs[t]; __syncthreads();
    for (int off = 1; off < 256; off <<= 1) {
        float add = (t >= off) ? zc[t - off] : 0.f;
        __syncthreads();
        zc[t] += add;
        __syncthreads();
    }
    if (t == 0) kz_s = 0;
    __syncthreads();
    int sup = (1.f + (float)(t + 1) * zs[t] > zc[t]) ? 1 : 0;
    atomicAdd(&kz_s, sup);
    __syncthreads();
    int kz = kz_s;
    float tau = (zc[kz - 1] - 1.f) / (float)kz;
    float m = fmaxf(z - tau, 0.f);
    mask[(size_t)row * 256 + t] = m;
    prior[(size_t)row * 256 + t] = pr * (GAMMA_C - m);
    red[t] = m * __logf(m + EPS_M);
    __syncthreads();
    for (int off = 128; off > 0; off >>= 1) {
        if (t < off) red[t] += red[t + off];
        __syncthreads();
    }
    if (t == 0) atomicAdd(loss_acc, red[0]);
}

// ---------------- rank-1 self-attention, one block per batch row ----------------
// logits are rank-1 (q_i * k_j), so row max is q_i*kmax (q_i>=0) or q_i*kmin.
// Phase 1: two block reductions give kmax/kmin.
// Phase 2: thread t owns output row i=t, accumulates sum/exp-weighted v with
//          zero syncs (LDS broadcast reads), writes y directly.
// Phase 3: streams amaps with coalesced stores (thread t = column t), one exp
//          per element, no reductions. Bandwidth-bound by design (~805 MB out).
__global__ void self_attn_kernel(const float* __restrict__ x,
                                 const float* __restrict__ in_w, const float* __restrict__ in_b,
                                 const float* __restrict__ out_w, const float* __restrict__ out_b,
                                 int s, float* __restrict__ y, float* __restrict__ amap) {
    int row = blockIdx.x, t = threadIdx.x;
    __shared__ float qq[256], kk[256], vv[256];
    __shared__ float mi[256], si[256];    // per-row softmax max and 1/sum
    __shared__ float red[256];
    float w0 = in_w[s * 3 + 0], w1 = in_w[s * 3 + 1], w2 = in_w[s * 3 + 2];
    float b0 = in_b[s * 3 + 0], b1 = in_b[s * 3 + 1], b2 = in_b[s * 3 + 2];
    float ow = out_w[s], ob = out_b[s];
    float xv = x[(size_t)row * 256 + t];
    float q = xv * w0 + b0;
    float k = xv * w1 + b1;
    float v = xv * w2 + b2;
    qq[t] = q; kk[t] = k; vv[t] = v;
    // kmax reduction
    red[t] = k; __syncthreads();
    for (int off = 128; off > 0; off >>= 1) {
        if (t < off) red[t] = fmaxf(red[t], red[t + off]);
        __syncthreads();
    }
    float kmax = red[0]; __syncthreads();
    // kmin reduction
    red[t] = k; __syncthreads();
    for (int off = 128; off > 0; off >>= 1) {
        if (t < off) red[t] = fminf(red[t], red[t + off]);
        __syncthreads();
    }
    float kmin = red[0]; __syncthreads();

    // thread t owns row i = t: exact row max via rank-1 structure
    float m = (q >= 0.f) ? q * kmax : q * kmin;
    float ssum = 0.f, yacc = 0.f;
    for (int j = 0; j < 256; ++j) {
        float e = __expf(q * kk[j] - m);
        ssum += e;
        yacc += e * vv[j];
    }
    float inv = 1.f / ssum;
    mi[t] = m; si[t] = inv;
    y[(size_t)row * 256 + t] = yacc * inv * ow + ob;
    __syncthreads();

    // stream attn map: thread t writes column t of every row i (coalesced b32)
    float* amr = amap + (size_t)row * 256 * 256;
    for (int i = 0; i < 256; ++i) {
        float a = __expf(qq[i] * k - mi[i]) * si[i];
        amr[(size_t)i * 256 + t] = a;
    }
}

extern "C" void kernel_launch(void* const* d_in, const int* in_sizes, int n_in,
                              void* d_out, int out_size, void* d_ws, size_t ws_size,
                              hipStream_t stream) {
    const float* x         = (const float*)d_in[0];
    const float* bemv      = (const float*)d_in[1];
    const float* gmat      = (const float*)d_in[2];
    const float* bn_w      = (const float*)d_in[3];
    const float* bn_b      = (const float*)d_in[4];
    const float* init_fc0  = (const float*)d_in[5];
    const float* init_bnw0 = (const float*)d_in[6];
    const float* init_bnb0 = (const float*)d_in[7];
    const float* init_fc1  = (const float*)d_in[8];
    const float* init_bnw1 = (const float*)d_in[9];
    const float* init_bnb1 = (const float*)d_in[10];
    const float* step_fc0  = (const float*)d_in[11];
    const float* step_bnw0 = (const float*)d_in[12];
    const float* step_bnb0 = (const float*)d_in[13];
    const float* step_fc1  = (const float*)d_in[14];
    const float* step_bnw1 = (const float*)d_in[15];
    const float* step_bnb1 = (const float*)d_in[16];
    const float* att_fc    = (const float*)d_in[17];
    const float* att_bnw   = (const float*)d_in[18];
    const float* att_bnb   = (const float*)d_in[19];
    const float* sa_in_w   = (const float*)d_in[20];
    const float* sa_in_b   = (const float*)d_in[21];
    const float* sa_out_w  = (const float*)d_in[22];
    const float* sa_out_b  = (const float*)d_in[23];

    float* out = (float*)d_out;
    float* steps_out = out;                        // 3*1024*64
    float* mloss_out = out + 196608;               // 1
    float* amaps_out = out + 196609;               // 3*1024*256*256

    // workspace layout (floats)
    float* W    = (float*)d_ws;
    float* xn   = W;                 // 262144
    float* xcat = xn   + 262144;     // 524288
    float* h    = xcat + 524288;     // 262144 (GEMM output buffer)
    float* glu0 = h    + 262144;     // 131072
    float* glu1 = glu0 + 131072;     // 131072
    float* feat = glu1 + 131072;     // 131072
    float* att  = feat + 131072;     // 65536
    float* prior= att  + 65536;      // 262144
    float* mask = prior+ 262144;     // 262144
    float* abuf = mask + 262144;     // 262144 (gbn(att gemm) output)
    float* xm   = abuf + 262144;     // 262144 (masked features)
    float* yb   = xm   + 262144;     // 262144 (self-attn output)
    float* loss = yb   + 262144;     // 1

    dim3 gG(GEMM_M / 16, GEMM_N / 64), bG(128);

    // xn = bn(x); x_cat = [xn, bemv]
    bn_kernel<<<256, 256, 0, stream>>>(x, bn_w, bn_b, xn);
    concat_kernel<<<(B_N * 512) / 256, 256, 0, stream>>>(xn, bemv, xcat);

    // initial feature transformer
    gemm_wmma<true><<<gG, bG, 0, stream>>>(xcat, init_fc0, h, D_CAT);
    gbn_glu_kernel<<<B_N / VBS, 256, 0, stream>>>(h, init_bnw0, init_bnb0, glu0);
    gemm_wmma<true><<<gG, bG, 0, stream>>>(glu0, init_fc1, h, FO);
    gbn_glu_kernel<<<B_N / VBS, 256, 0, stream>>>(h, init_bnw1, init_bnb1, glu1);
    residual_kernel<<<(B_N * FO) / 256, 256, 0, stream>>>(glu0, glu1, feat);
    split_feat_kernel<<<(B_N * FO) / 256, 256, 0, stream>>>(feat, steps_out, att, 0);

    fill_kernel<<<(B_N * AD) / 256, 256, 0, stream>>>(prior, 1.0f, B_N * AD);
    fill_kernel<<<1, 256, 0, stream>>>(loss, 0.0f, 1);

    for (int s = 0; s < N_STEPS; ++s) {
        // a = gbn(att @ att_fc[s].T)
        gemm_wmma<true><<<gG, bG, 0, stream>>>(att, att_fc + (size_t)s * AD * N_AA, h, N_AA);
        gbn_kernel<<<B_N / VBS, 256, 0, stream>>>(h, att_bnw + s * AD, att_bnb + s * AD, abuf);
        // sparsemax + entropy loss + prior update
        sparsemax_kernel<<<B_N, 256, 0, stream>>>(abuf, prior, mask, loss);
        // x_masked = (mask @ gmat) * xn
        gemm_wmma<false><<<gG, bG, 0, stream>>>(mask, gmat, h, AD);
        mul_kernel<<<(B_N * F_INN) / 256, 256, 0, stream>>>(h, xn, xm);
        // self-attention (writes amaps directly into d_out)
        self_attn_kernel<<<B_N, 256, 0, stream>>>(
            xm, sa_in_w, sa_in_b, sa_out_w, sa_out_b, s, yb,
            amaps_out + (size_t)s * B_N * 256 * 256);
        // step feature transformer
        concat_kernel<<<(B_N * 512) / 256, 256, 0, stream>>>(yb, bemv, xcat);
        gemm_wmma<true><<<gG, bG, 0, stream>>>(xcat, step_fc0 + (size_t)s * 256 * D_CAT, h, D_CAT);
        gbn_glu_kernel<<<B_N / VBS, 256, 0, stream>>>(h, step_bnw0 + s * 256, step_bnb0 + s * 256, glu0);
        gemm_wmma<true><<<gG, bG, 0, stream>>>(glu0, step_fc1 + (size_t)s * 256 * FO, h, FO);
        gbn_glu_kernel<<<B_N / VBS, 256, 0, stream>>>(h, step_bnw1 + s * 256, step_bnb1 + s * 256, glu1);
        residual_kernel<<<(B_N * FO) / 256, 256, 0, stream>>>(glu0, glu1, feat);
        split_feat_kernel<<<(B_N * FO) / 256, 256, 0, stream>>>(
            feat, steps_out + (size_t)s * B_N * N_DD, att, 1);
    }
    finalize_loss_kernel<<<1, 1, 0, stream>>>(loss, mloss_out);
}